// SINTokInputEmbCONCAT_77936476553915
// MI455X (gfx1250) — compile-verified
//
#include <hip/hip_runtime.h>
#include <hip/hip_bf16.h>
#include <math.h>

// ---------------------------------------------------------------------------
// SINTokInputEmb + LayerNorm, MI455X (gfx1250, wave32)
//
// out[b,s,:] = LN( word_emb[id] + pe[s] + type_emb[tt] + concat(g,g,g) ) * w + b
//   pe   : 768-dim sinusoid of position s
//   g    : 256-dim sinusoid of para_pos (tok_struct_vec[...,0])
//
// Memory-bound (~200 MB moved, ~9 us floor at 23.3 TB/s). One wave32 per
// token; lane owns 6 float4 chunks (24 of 768 floats). Sin tables are
// computed analytically in registers (freqs hoisted out of the token loop),
// LayerNorm is a wave32 __shfl_xor butterfly, output uses non-temporal b128
// stores to keep the 93.7MB word_emb table resident in the 192MB L2, and the
// next token's gather row is warmed with global_prefetch_b8.
// ---------------------------------------------------------------------------

typedef __attribute__((ext_vector_type(4))) float v4f;

#define HID       768
#define SEQ       4096
#define CHUNKS    6            // 6 float4 per lane * 32 lanes * 4 = 768
#define LN10000   9.210340371976184f

__global__ __launch_bounds__(256) void
emb_ln_kernel(const int*   __restrict__ input_ids,    // [B*S]
              const int*   __restrict__ tok_struct,   // [B*S*3]
              const int*   __restrict__ token_type,   // [B*S]
              const float* __restrict__ word_emb,     // [VOCAB*768]
              const float* __restrict__ type_emb,     // [2*768]
              const float* __restrict__ ln_w,         // [768]
              const float* __restrict__ ln_b,         // [768]
              float*       __restrict__ out,          // [B*S*768]
              int n_tokens)
{
    const int lane   = threadIdx.x & 31;
    const int waveIB = threadIdx.x >> 5;
    const int wavesPerBlock = blockDim.x >> 5;
    const int wave   = blockIdx.x * wavesPerBlock + waveIB;
    const int nWaves = gridDim.x * wavesPerBlock;

    // ---- loop-invariant per-lane state: inverse frequencies + ln params ----
    const float kpe = LN10000 / 768.0f;   // pe  table, dim 768
    const float khs = LN10000 / 256.0f;   // hs_pe table, dim 256
    float fpe0[CHUNKS], fpe1[CHUNKS], fhs0[CHUNKS], fhs1[CHUNKS];
    v4f   lw[CHUNKS], lb[CHUNKS];
#pragma unroll
    for (int j = 0; j < CHUNKS; ++j) {
        const int d  = 4 * (j * 32 + lane);      // even, multiple of 4
        fpe0[j] = __expf(-(float)d        * kpe);
        fpe1[j] = __expf(-(float)(d + 2)  * kpe);
        fhs0[j] = __expf(-(float)(d & 255)        * khs);
        fhs1[j] = __expf(-(float)((d + 2) & 255)  * khs);
        lw[j] = *(const v4f*)(ln_w + d);
        lb[j] = *(const v4f*)(ln_b + d);
    }

    for (int t = wave; t < n_tokens; t += nWaves) {
        // ---- warm L2/L0 for the NEXT iteration's gathered word row --------
        const int tn = t + nWaves;
        if (tn < n_tokens) {
            const int nid = input_ids[tn];                       // s_load (uniform)
            const char* prow = (const char*)(word_emb + (size_t)nid * HID);
            if (lane < 24)                                       // 24 x 128B = 3KB row
                __builtin_prefetch(prow + (size_t)lane * 128, 0, 3); // global_prefetch_b8
        }

        const int   wid = input_ids[t];
        const int   tt  = token_type[t];
        const int   pos = t & (SEQ - 1);                 // s  (S = 4096)
        const int   pp  = tok_struct[3 * (size_t)t];     // para_pos
        const float sf  = (float)pos;
        const float pf  = (float)pp;

        const v4f* wrow = (const v4f*)(word_emb + (size_t)wid * HID);
        const v4f* trow = (const v4f*)(type_emb + (size_t)tt  * HID);

        v4f   v[CHUNKS];
        float sum = 0.0f, ssq = 0.0f;
#pragma unroll
        for (int j = 0; j < CHUNKS; ++j) {
            const int f  = j * 32 + lane;                // float4 index (coalesced)
            const v4f w  = wrow[f];                      // global_load_b128
            const v4f ty = trow[f];                      // L0-resident (2 rows)
            const float a0 = sf * fpe0[j], a1 = sf * fpe1[j];   // pe angles
            const float b0 = pf * fhs0[j], b1 = pf * fhs1[j];   // struct angles
            v4f e;
            e.x = w.x + ty.x + __sinf(a0) + __sinf(b0);  // even dims -> sin
            e.y = w.y + ty.y + __cosf(a0) + __cosf(b0);  // odd  dims -> cos
            e.z = w.z + ty.z + __sinf(a1) + __sinf(b1);
            e.w = w.w + ty.w + __cosf(a1) + __cosf(b1);
            v[j] = e;
            sum += e.x + e.y + e.z + e.w;
            ssq += e.x * e.x + e.y * e.y + e.z * e.z + e.w * e.w;
        }

        // ---- wave32 butterfly reduction (no LDS alloc, no barrier) --------
#pragma unroll
        for (int off = 16; off > 0; off >>= 1) {
            sum += __shfl_xor(sum, off, 32);
            ssq += __shfl_xor(ssq, off, 32);
        }
        const float inv = 1.0f / 768.0f;
        const float mu  = sum * inv;
        const float var = ssq * inv - mu * mu;
        const float rs  = rsqrtf(var + 1e-12f);

        // ---- normalize + affine, streamed out non-temporally --------------
        v4f* orow = (v4f*)(out + (size_t)t * HID);
#pragma unroll
        for (int j = 0; j < CHUNKS; ++j) {
            const int f = j * 32 + lane;
            const v4f e = v[j];
            v4f o;
            o.x = (e.x - mu) * rs * lw[j].x + lb[j].x;
            o.y = (e.y - mu) * rs * lw[j].y + lb[j].y;
            o.z = (e.z - mu) * rs * lw[j].z + lb[j].z;
            o.w = (e.w - mu) * rs * lw[j].w + lb[j].w;
            __builtin_nontemporal_store(o, &orow[f]);    // b128 store, NT hint
        }
    }
}

extern "C" void kernel_launch(void* const* d_in, const int* in_sizes, int n_in,
                              void* d_out, int out_size, void* d_ws, size_t ws_size,
                              hipStream_t stream) {
    const int*   input_ids  = (const int*)  d_in[0];   // (B,S)
    const int*   tok_struct = (const int*)  d_in[1];   // (B,S,3)
    /* d_in[2] = sent_struct_vec: unused by the reference */
    const int*   token_type = (const int*)  d_in[3];   // (B,S)
    const float* word_emb   = (const float*)d_in[4];   // (VOCAB,768)
    const float* type_emb   = (const float*)d_in[5];   // (2,768)
    const float* ln_w       = (const float*)d_in[6];   // (768,)
    const float* ln_b       = (const float*)d_in[7];   // (768,)
    float*       out        = (float*)      d_out;

    const int n_tokens = in_sizes[0];                  // B*S = 32768

    // 2048 blocks x 8 waves = 16384 wave32s, grid-stride (2 tokens/wave)
    // -> tens of thousands of independent gather streams for the 23.3 TB/s HBM.
    dim3 grid(2048), block(256);
    hipLaunchKernelGGL(emb_ln_kernel, grid, block, 0, stream,
                       input_ids, tok_struct, token_type,
                       word_emb, type_emb, ln_w, ln_b, out, n_tokens);
}